// QueryAndGroupCnt_31576599560761
// MI455X (gfx1250) — compile-verified
//
#include <hip/hip_runtime.h>

typedef __attribute__((ext_vector_type(2))) float v2f;
typedef __attribute__((ext_vector_type(8))) float v8f;

constexpr int   B_    = 4;
constexpr int   N_    = 16384;
constexpr int   NP_   = 2048;
constexpr int   C_    = 64;
constexpr int   NS_   = 32;
constexpr float RAD2  = 0.1f * 0.1f;
constexpr int   NCH   = N_ / 16;           // 1024 chunks of 16 points
constexpr int   NTILE = B_ * (NP_ / 16);   // 512 query tiles of 16 queries
constexpr int   NSEG  = 4;                 // waves per tile (N split 4 ways)
constexpr int   NCHSEG = NCH / NSEG;       // 256 chunks per segment

// ---------------------------------------------------------------------------
// Kernel A: ball query via V_WMMA_F32_16X16X4_F32, N split across 4 waves.
//   A row m  = (-2qx, -2qy, -2qz, 1)
//   B col n  = ( px,   py,   pz,  |p|^2)
//   D[m][n]  = |p|^2 - 2 q.p            (in-radius  <=>  D < r^2 - |q|^2)
// Each of the 4 waves in a block scans one quarter of the cloud for the same
// 16 queries, appending in-order hits (capped at 32) to a per-segment LDS
// list via ds_store. After a barrier, 128 threads merge the 4 ordered lists
// per row (prefix-sum slot resolution) and emit the final 32 indices per
// query as coalesced 128B lines, plus the clamped count.
// Loads are pipelined distance-1 (unconditional, clamped address) and a
// global_prefetch_b8 slab prefetch runs 8 chunks ahead.
// ---------------------------------------------------------------------------
__global__ __launch_bounds__(128)
void ballquery16_wmma(const float* __restrict__ xyz,
                      const float* __restrict__ new_xyz,
                      int* __restrict__ cnt_out,
                      int* __restrict__ idx_out)
{
    __shared__ int lidx[NSEG][16][NS_];   // per-segment ordered hit lists
    __shared__ int lcnt[NSEG][16];        // per-segment hit counts (clamped)

    const int lane = threadIdx.x & 31;
    const int w    = threadIdx.x >> 5;    // segment id 0..3
    const int tile = blockIdx.x;          // one tile per block (grid == NTILE)
    const int b  = tile / (NP_ / 16);
    const int qb = (tile % (NP_ / 16)) * 16;

    // lanes l and l+16 both hold query m = l & 15 (A-matrix K split by half-wave)
    const int m = lane & 15;
    const float* q = new_xyz + (size_t)(b * NP_ + qb + m) * 3;
    const float qx = q[0], qy = q[1], qz = q[2];
    const float qn = qx * qx + qy * qy + qz * qz;

    // A (16x4 f32): lanes 0-15 hold K=0,1 ; lanes 16-31 hold K=2,3.
    v2f A;
    A.x = (lane < 16) ? (-2.0f * qx) : (-2.0f * qz);
    A.y = (lane < 16) ? (-2.0f * qy) : 1.0f;

    // Per-row threshold r^2 - |q|^2: D VGPR v holds row v (lanes 0-15) and
    // row v+8 (lanes 16-31).
    float thr8[8];
#pragma unroll
    for (int v = 0; v < 8; ++v)
        thr8[v] = RAD2 - __shfl(qn, v + ((lane >> 4) << 3), 32);

    int cnt = 0;
    const float* pbase = xyz + (size_t)b * N_ * 3;
    const int ch0 = w * NCHSEG;
    const int ch1 = ch0 + NCHSEG;

    // software pipeline: prime first chunk of this segment
    const float* p0 = pbase + (size_t)(ch0 * 16 + m) * 3;
    float px = p0[0], py = p0[1], pz = p0[2];

    for (int ch = ch0; ch < ch1; ++ch) {
        // slab prefetch 8 chunks ahead: 32 lanes x 48B stride cover the
        // 1536B (8-chunk) slab -> global_prefetch_b8 touches every line.
        if (((ch - ch0) & 7) == 0) {
            int pfch = ch + 8;
            if (pfch > NCH - 8) pfch = NCH - 8;
            const char* pf = (const char*)(pbase + (size_t)pfch * 16 * 3) + lane * 48;
            __builtin_prefetch(pf, 0, 3);
        }

        // unconditional distance-1 prefetch (clamped) -> loadcnt<=1 pipelining
        const int chn = (ch + 1 < ch1) ? (ch + 1) : ch;
        const float* pnx = pbase + (size_t)(chn * 16 + m) * 3;
        const float nxp = pnx[0], nyp = pnx[1], nzp = pnx[2];

        const float pn = px * px + py * py + pz * pz;

        // B (4x16 f32): lanes 0-15 rows K=0,1 ; lanes 16-31 rows K=2,3.
        v2f Bm;
        Bm.x = (lane < 16) ? px : pz;
        Bm.y = (lane < 16) ? py : pn;

        v8f Cz = {};
        v8f D = __builtin_amdgcn_wmma_f32_16x16x4_f32(
            false, A, false, Bm, (short)0, Cz, false, false);

        // predicate tile -> per-row 16-bit masks via 8 ballots (the v_cmp IS
        // the ballot); distribute to owner lanes with 8 selects.
        unsigned fullmask = 0u;
#pragma unroll
        for (int v = 0; v < 8; ++v) {
            unsigned mv = __builtin_amdgcn_ballot_w32(D[v] < thr8[v]);
            if ((lane & 7) == v) fullmask = mv;   // rows v (lo16) / v+8 (hi16)
        }

        if (lane < 16) {
            unsigned rowmask = (lane < 8) ? (fullmask & 0xFFFFu) : (fullmask >> 16);
            while (rowmask && cnt < NS_) {
                int bit = __builtin_ctz(rowmask);
                lidx[w][lane][cnt] = ch * 16 + bit;   // ds_store, no vmem stall
                ++cnt;
                rowmask &= rowmask - 1;
            }
        }

        // wave-uniform early exit: all 16 rows saturated in this segment
        if (__builtin_amdgcn_ballot_w32(lane < 16 && cnt < NS_) == 0u)
            break;

        px = nxp; py = nyp; pz = nzp;
    }

    if (lane < 16) lcnt[w][lane] = cnt;
    __syncthreads();

    // ---- merge the 4 ordered per-segment lists: 512 slots over 128 threads.
    // Output writes are coalesced (consecutive j across lanes).
#pragma unroll
    for (int k = 0; k < 4; ++k) {
        const int slot = threadIdx.x + k * 128;   // 0..511
        const int row  = slot >> 5;               // query row 0..15
        const int j    = slot & 31;               // sample slot 0..31

        const int c0 = lcnt[0][row], c1 = lcnt[1][row];
        const int c2 = lcnt[2][row], c3 = lcnt[3][row];
        const int t01 = c0 + c1, t012 = t01 + c2, tot = t012 + c3;

        int val;
        if      (j < c0)   val = lidx[0][row][j];
        else if (j < t01)  val = lidx[1][row][j - c0];
        else if (j < t012) val = lidx[2][row][j - t01];
        else if (j < tot)  val = lidx[3][row][j - t012];
        else {
            int fill = 0;                          // first hit overall, 0 if none
            if      (c0) fill = lidx[0][row][0];
            else if (c1) fill = lidx[1][row][0];
            else if (c2) fill = lidx[2][row][0];
            else if (c3) fill = lidx[3][row][0];
            val = fill;
        }
        idx_out[(size_t)(b * NP_ + qb + row) * NS_ + j] = val;
        if (j == 0)
            cnt_out[b * NP_ + qb + row] = (tot < NS_) ? tot : NS_;
    }
}

// ---------------------------------------------------------------------------
// Kernel B: gather/group. One block per (b, query); lane = sample index so the
// 32 stores per channel form one coalesced 128B line. All gathers L2-resident.
// out[b][c][q][s] : c in [0,3) -> centered xyz, c in [3,67) -> features[c-3].
// ---------------------------------------------------------------------------
__global__ __launch_bounds__(256)
void group_features(const float* __restrict__ xyz,
                    const float* __restrict__ new_xyz,
                    const float* __restrict__ feats,
                    const int* __restrict__ idx_in,
                    float* __restrict__ out)
{
    const int bq = blockIdx.x;           // 0 .. B*NP-1
    const int b  = bq / NP_;
    const int qp = bq - b * NP_;
    const int s  = threadIdx.x & 31;
    const int cr = threadIdx.x >> 5;     // channel phase 0..7 (uniform per wave)

    const int pi = idx_in[(size_t)bq * NS_ + s];

    const float* pc = xyz + ((size_t)b * N_ + pi) * 3;
    const float* qc = new_xyz + (size_t)bq * 3;
    const float* fb = feats + (size_t)b * C_ * N_;

    for (int c = cr; c < 3 + C_; c += 8) {
        float val;
        if (c < 3) val = pc[c] - qc[c];
        else       val = fb[(size_t)(c - 3) * N_ + pi];
        out[(((size_t)b * (3 + C_) + c) * NP_ + qp) * NS_ + s] = val;
    }
}

// ---------------------------------------------------------------------------
extern "C" void kernel_launch(void* const* d_in, const int* in_sizes, int n_in,
                              void* d_out, int out_size, void* d_ws, size_t ws_size,
                              hipStream_t stream)
{
    const float* xyz     = (const float*)d_in[0];
    const float* new_xyz = (const float*)d_in[1];
    const float* feats   = (const float*)d_in[2];

    int*   cnt_out  = (int*)d_out;                       // idx_cnt, int32 bits
    float* feat_out = (float*)d_out + (size_t)B_ * NP_;  // new_features
    int*   idx_ws   = (int*)d_ws;                        // B*NP*NS ints = 1 MB

    dim3 blkA(128), grdA(NTILE);
    ballquery16_wmma<<<grdA, blkA, 0, stream>>>(xyz, new_xyz, cnt_out, idx_ws);

    dim3 blkB(256), grdB(B_ * NP_);
    group_features<<<grdB, blkB, 0, stream>>>(xyz, new_xyz, feats, idx_ws, feat_out);
}